// CausalBoW_12223476924918
// MI455X (gfx1250) — compile-verified
//
#include <hip/hip_runtime.h>

// ---------------------------------------------------------------------------
// Causal uniform attention == running causal mean over T.
// x: (B=8, T=4096, C=1024) f32, y same shape.
// Decoupled scan: pass1 chunk sums (WMMA ones-matrix reduction),
// pass2 carry + in-chunk scan + scale (async global->LDS streaming).
// Roofline: 256 MiB HBM traffic floor ~11us @ 23.3 TB/s; x (128 MiB) fits in
// the 192 MB L2, so pass-2's re-read of x is L2-resident.
// ---------------------------------------------------------------------------

#define AS1 __attribute__((address_space(1)))
#define AS3 __attribute__((address_space(3)))

typedef __attribute__((ext_vector_type(2))) float v2f;
typedef __attribute__((ext_vector_type(8))) float v8f;
typedef int v4i __attribute__((vector_size(16)));

#ifndef __has_builtin
#define __has_builtin(x) 0
#endif

#if __has_builtin(__builtin_amdgcn_wmma_f32_16x16x4_f32)
#define HAVE_WMMA 1
#endif
#if __has_builtin(__builtin_amdgcn_global_load_async_to_lds_b128)
#define HAVE_ASYNC 1
#endif

constexpr int kB = 8;
constexpr int kT = 4096;
constexpr int kC = 1024;
constexpr int kChunk = 64;               // timesteps per chunk
constexpr int kNChunk = kT / kChunk;     // 64 chunks
constexpr int kTileT = 8;                // rows per async double-buffer tile
constexpr int kNTile = kChunk / kTileT;  // 8 tiles per chunk

// ---------------------------------------------------------------------------
// Kernel 1: partial sums per (b, chunk, c).
// Each wave owns 16 channels. WMMA trick: A = ones(16x4), so
// D[m][n] = sum_k B[k][n] + C[m][n]  -> every row of D is the column sum.
// Chaining C across the chunk accumulates 64 timesteps in 16 WMMAs.
// Lane l (<16) supplies B values for channel c0+l, rows {tt,tt+1};
// lane l (>=16) supplies channel c0+l-16, rows {tt+2,tt+3}. With ones-A the
// K-slot assignment is irrelevant; column<->lane mapping is consistent B->D.
// ---------------------------------------------------------------------------
__global__ __launch_bounds__(256) void k_chunk_sums(const float* __restrict__ x,
                                                    float* __restrict__ partials) {
  const int wave  = threadIdx.x >> 5;
  const int lane  = threadIdx.x & 31;
  const int n     = lane & 15;
  const int khalf = lane >> 4;  // 0: t rows {0,1}; 1: t rows {2,3}
  const int c0    = blockIdx.x * 128 + wave * 16;
  const int chunk = blockIdx.y;
  const int b     = blockIdx.z;

  const float* p0 =
      x + ((size_t)b * kT + (size_t)chunk * kChunk + (size_t)(khalf * 2)) * kC + (c0 + n);

#if defined(HAVE_WMMA)
  v8f acc = {0.f, 0.f, 0.f, 0.f, 0.f, 0.f, 0.f, 0.f};
  v2f ones;
  ones.x = 1.0f;
  ones.y = 1.0f;
#pragma unroll 4
  for (int tt = 0; tt < kChunk; tt += 4) {
    v2f bm;
    bm.x = p0[(size_t)tt * kC];        // t = tt + khalf*2
    bm.y = p0[(size_t)(tt + 1) * kC];  // t = tt + khalf*2 + 1
    acc = __builtin_amdgcn_wmma_f32_16x16x4_f32(false, ones, false, bm, (short)0, acc,
                                                false, false);
  }
  if (lane < 16)
    partials[((size_t)b * kNChunk + chunk) * kC + c0 + lane] = acc[0];
#else
  // VALU fallback: half-lanes split the 4-row groups, combine across halves.
  float s = 0.f;
  for (int tt = 0; tt < kChunk; tt += 4)
    s += p0[(size_t)tt * kC] + p0[(size_t)(tt + 1) * kC];
  s += __shfl_xor(s, 16, 32);
  if (lane < 16)
    partials[((size_t)b * kNChunk + chunk) * kC + c0 + lane] = s;
#endif
}

// ---------------------------------------------------------------------------
// Kernel 2: per-(b,chunk) block. Recompute exclusive carry from partials
// (L2-resident, tiny), then stream the 64-row slab with double-buffered
// async global->LDS loads, serial in-chunk prefix per lane (float4/thread),
// scale by 1/(t+1), store.
// ---------------------------------------------------------------------------
__global__ __launch_bounds__(256) void k_scan_apply(const float* __restrict__ x,
                                                    const float* __restrict__ partials,
                                                    float* __restrict__ y) {
  const int tid   = threadIdx.x;
  const int chunk = blockIdx.x;
  const int b     = blockIdx.y;
  const int c4    = tid * 4;  // 256 threads * float4 = full 1024-channel row

  // Exclusive carry for this chunk.
  float cx = 0.f, cy = 0.f, cz = 0.f, cw = 0.f;
  for (int k = 0; k < chunk; ++k) {
    const float* pp = partials + ((size_t)b * kNChunk + k) * kC + c4;
    cx += pp[0];
    cy += pp[1];
    cz += pp[2];
    cw += pp[3];
  }

  const int t0    = chunk * kChunk;
  const float* xb = x + ((size_t)b * kT + t0) * kC;
  float* yb       = y + ((size_t)b * kT + t0) * kC;

#if defined(HAVE_ASYNC)
  __shared__ float lds[2][kTileT][kC];  // 64 KB, double buffered

  auto issue_tile = [&](int tile) {
    const int buf = tile & 1;
#pragma unroll
    for (int r = 0; r < kTileT; ++r) {
      const float* g = xb + (size_t)(tile * kTileT + r) * kC + c4;
      __builtin_amdgcn_global_load_async_to_lds_b128(
          (AS1 v4i*)g, (AS3 v4i*)(&lds[buf][r][c4]), 0, 0);
    }
  };

  issue_tile(0);
  for (int tile = 0; tile < kNTile; ++tile) {
    if (tile + 1 < kNTile) {
      issue_tile(tile + 1);  // prefetch next tile, then wait for current one
#if __has_builtin(__builtin_amdgcn_s_wait_asynccnt)
      __builtin_amdgcn_s_wait_asynccnt(kTileT);
#else
      asm volatile("s_wait_asynccnt 8" ::: "memory");
#endif
    } else {
#if __has_builtin(__builtin_amdgcn_s_wait_asynccnt)
      __builtin_amdgcn_s_wait_asynccnt(0);
#else
      asm volatile("s_wait_asynccnt 0" ::: "memory");
#endif
    }
    const int buf = tile & 1;
#pragma unroll
    for (int r = 0; r < kTileT; ++r) {
      const int tl   = tile * kTileT + r;
      const float4 v = *reinterpret_cast<const float4*>(&lds[buf][r][c4]);
      cx += v.x;
      cy += v.y;
      cz += v.z;
      cw += v.w;
      const float s = 1.0f / (float)(t0 + tl + 1);
      float4 o;
      o.x = cx * s;
      o.y = cy * s;
      o.z = cz * s;
      o.w = cw * s;
      *reinterpret_cast<float4*>(&yb[(size_t)tl * kC + c4]) = o;
    }
  }
#else
  // Fallback: direct coalesced float4 streaming (L2-resident after pass 1).
  for (int r = 0; r < kChunk; ++r) {
    const float4 v = *reinterpret_cast<const float4*>(&xb[(size_t)r * kC + c4]);
    cx += v.x;
    cy += v.y;
    cz += v.z;
    cw += v.w;
    const float s = 1.0f / (float)(t0 + r + 1);
    float4 o;
    o.x = cx * s;
    o.y = cy * s;
    o.z = cz * s;
    o.w = cw * s;
    *reinterpret_cast<float4*>(&yb[(size_t)r * kC + c4]) = o;
  }
#endif
}

// ---------------------------------------------------------------------------
extern "C" void kernel_launch(void* const* d_in, const int* in_sizes, int n_in,
                              void* d_out, int out_size, void* d_ws, size_t ws_size,
                              hipStream_t stream) {
  (void)in_sizes;
  (void)n_in;
  (void)out_size;
  (void)ws_size;

  const float* x  = (const float*)d_in[0];
  float* y        = (float*)d_out;
  float* partials = (float*)d_ws;  // kB * kNChunk * kC floats = 2 MiB

  // Pass 1: per-chunk column sums (WMMA ones-reduction).
  k_chunk_sums<<<dim3(kC / 128, kNChunk, kB), 256, 0, stream>>>(x, partials);
  // Pass 2: carry + in-chunk scan + scale (async LDS streaming).
  k_scan_apply<<<dim3(kNChunk, kB), 256, 0, stream>>>(x, partials, y);
}